// AttnDecoderRNN_81226421502459
// MI455X (gfx1250) — compile-verified
//
#include <hip/hip_runtime.h>
#include <math.h>

#define H 256
#define L 20
#define V 29
#define LAT 264

typedef __attribute__((ext_vector_type(2))) float v2f;
typedef __attribute__((ext_vector_type(8))) float v8f;

// ---------------------------------------------------------------------------
// One 16-output GEMV tile via V_WMMA_F32_16X16X4_F32 (pure f32 matrix op).
//   out[tile*16 + m] = bias[...] + sum_k W[m_glob][k] * x[k]
// Mapping: A = 16x4 weight tile (lane L holds row L&15; VGPR0/1 hold
// K = {0,1} for lanes 0-15 and K = {2,3} for lanes 16-31 per the ISA A-layout).
// B = 4x16 with every column equal to x[k..k+3], so every column of D holds
// the same 16 partial outputs; we read column 0 (lanes 0 and 16).
// Whole wave executes the WMMA loop uniformly (EXEC all-ones requirement).
// ---------------------------------------------------------------------------
__device__ __forceinline__ void wmma_gemv_tile(
    const float* __restrict__ W,     // [Nvalid, K] row-major
    const float* __restrict__ x,     // length K (LDS)
    const float* __restrict__ bias,  // length Nvalid
    float* __restrict__ out,         // base of output array (absolute indexing)
    int K, int Nvalid, int tile, bool do_relu)
{
    const int lane = threadIdx.x & 31;
    const int half = lane >> 4;       // 0: K+{0,1}, 1: K+{2,3}
    const int row  = lane & 15;
    int m = tile * 16 + row;
    if (m > Nvalid - 1) m = Nvalid - 1;          // clamp; masked at store
    const float* wrow = W + (size_t)m * (size_t)K;

    v8f acc = (v8f){0.f, 0.f, 0.f, 0.f, 0.f, 0.f, 0.f, 0.f};
    for (int k = 0; k < K; k += 4) {
        const int kk = k + 2 * half;
        v2f a; a.x = wrow[kk]; a.y = wrow[kk + 1];
        v2f b; b.x = x[kk];    b.y = x[kk + 1];     // ds_load broadcast of x
        acc = __builtin_amdgcn_wmma_f32_16x16x4_f32(
            false, a, false, b, (short)0, acc, false, false);
    }

    // D layout: VGPR j, lanes 0-15 -> M=j (N=lane); lanes 16-31 -> M=j+8.
    // Column N=0 lives in lane 0 (M=0..7) and lane 16 (M=8..15).
    if (row == 0) {
        const int base = tile * 16 + half * 8;
#pragma unroll
        for (int j = 0; j < 8; ++j) {
            const int o = base + j;
            if (o < Nvalid) {
                float v = acc[j] + bias[o];
                out[o] = do_relu ? fmaxf(v, 0.f) : v;
            }
        }
    }
}

__device__ __forceinline__ float wave_max(float v) {
#pragma unroll
    for (int o = 16; o > 0; o >>= 1) v = fmaxf(v, __shfl_xor(v, o));
    return v;
}
__device__ __forceinline__ float wave_sum(float v) {
#pragma unroll
    for (int o = 16; o > 0; o >>= 1) v += __shfl_xor(v, o);
    return v;
}

// ---------------------------------------------------------------------------
// K1: embedding gather, latent2decoder, attention logits, softmax, context.
// Single workgroup (dependency-dense, tiny). 256 threads = 8 waves.
// ---------------------------------------------------------------------------
__global__ __launch_bounds__(256) void k1_front(
    const int* __restrict__ input_tensor, const float* __restrict__ hidden,
    const float* __restrict__ enc, const int* __restrict__ cond,
    const int* __restrict__ is_head,
    const float* __restrict__ emb,
    const float* __restrict__ w_l2d, const float* __restrict__ b_l2d,
    const float* __restrict__ w_attn, const float* __restrict__ b_attn,
    float* __restrict__ ws_hidden,   // [H]
    float* __restrict__ ws_xcomb,    // [2H] = [embedded | attn_applied]
    float* __restrict__ d_attn_out)  // d_out + 285, [L]
{
    __shared__ float s_eh[2 * H];    // [0:H)=embedded, [H:2H)=hidden'  (attn_in)
    __shared__ float s_lat[LAT];     // [hidden_in | onehot(8)]
    __shared__ float s_logits[32];
    __shared__ float s_aw[32];

    const int tid  = threadIdx.x;
    const int wave = tid >> 5;

    if (tid < H) {
        s_lat[tid] = hidden[tid];
        s_eh[tid]  = emb[input_tensor[0] * H + tid];
    }
    if (tid < 8) {
        const int c = (tid < 4) ? cond[0] : cond[1];
        const int p = (tid < 4) ? tid : tid - 4;
        s_lat[H + tid] = (p == c) ? 1.0f : 0.0f;
    }
    __syncthreads();

    if (is_head[0]) {                         // uniform scalar branch
        wmma_gemv_tile(w_l2d, s_lat, b_l2d, s_eh + H, LAT, H, wave,     false);
        wmma_gemv_tile(w_l2d, s_lat, b_l2d, s_eh + H, LAT, H, wave + 8, false);
    } else {
        if (tid < H) s_eh[H + tid] = hidden[tid];
    }
    __syncthreads();
    if (tid < H) ws_hidden[tid] = s_eh[H + tid];

    // attention logits: [emb|hidden'] (2H) @ w_attn.T -> L=20 (2 tiles)
    if (wave < 2)
        wmma_gemv_tile(w_attn, s_eh, b_attn, s_logits, 2 * H, L, wave, false);
    __syncthreads();

    if (wave == 0) {                          // softmax over 20 in one wave
        float v  = (tid < L) ? s_logits[tid] : -1e30f;
        float mx = wave_max(v);
        float e  = (tid < L) ? expf(v - mx) : 0.f;
        float sm = wave_sum(e);
        float w  = e / sm;
        if (tid < L) { s_aw[tid] = w; d_attn_out[tid] = w; }
    }
    __syncthreads();

    if (tid < H) {                            // context = aw @ encoder_outputs
        float s = 0.f;
#pragma unroll
        for (int l = 0; l < L; ++l) s = fmaf(s_aw[l], enc[l * H + tid], s);
        ws_xcomb[tid]     = s_eh[tid];        // embedded
        ws_xcomb[H + tid] = s;                // attn_applied
    }
}

// ---------------------------------------------------------------------------
// K2: comb GEMV (16 tiles, relu) + W_hh GEMV (48 tiles) in parallel across
// 64 one-wave workgroups — spreads the weight stream over many WGPs.
// ---------------------------------------------------------------------------
__global__ __launch_bounds__(32) void k2_comb_gh(
    const float* __restrict__ w_comb, const float* __restrict__ b_comb,
    const float* __restrict__ w_hh,   const float* __restrict__ b_hh,
    const float* __restrict__ ws_xcomb, const float* __restrict__ ws_hidden,
    float* __restrict__ ws_outp, float* __restrict__ ws_gh)
{
    __shared__ float s_x[2 * H];
    const int tid = threadIdx.x;
    const int b   = blockIdx.x;
    if (b < 16) {
        for (int i = tid; i < 2 * H; i += 32) s_x[i] = ws_xcomb[i];
        __syncthreads();
        wmma_gemv_tile(w_comb, s_x, b_comb, ws_outp, 2 * H, H, b, true);
    } else {
        for (int i = tid; i < H; i += 32) s_x[i] = ws_hidden[i];
        __syncthreads();
        wmma_gemv_tile(w_hh, s_x, b_hh, ws_gh, H, 3 * H, b - 16, false);
    }
}

// ---------------------------------------------------------------------------
// K3: W_ih GEMV (depends on comb output). 48 one-wave workgroups.
// ---------------------------------------------------------------------------
__global__ __launch_bounds__(32) void k3_gi(
    const float* __restrict__ w_ih, const float* __restrict__ b_ih,
    const float* __restrict__ ws_outp, float* __restrict__ ws_gi)
{
    __shared__ float s_x[H];
    const int tid = threadIdx.x;
    for (int i = tid; i < H; i += 32) s_x[i] = ws_outp[i];
    __syncthreads();
    wmma_gemv_tile(w_ih, s_x, b_ih, ws_gi, H, 3 * H, blockIdx.x, false);
}

// ---------------------------------------------------------------------------
// K4: GRU gate fusion, output GEMV (2 tiles), log_softmax. Single workgroup.
// ---------------------------------------------------------------------------
__global__ __launch_bounds__(256) void k4_tail(
    const float* __restrict__ ws_gi, const float* __restrict__ ws_gh,
    const float* __restrict__ ws_hidden,
    const float* __restrict__ w_out, const float* __restrict__ b_out,
    float* __restrict__ d_out)        // [0:V) logits, [V:V+H) h_new
{
    __shared__ float s_h[H];
    __shared__ float s_logit[32];
    const int tid  = threadIdx.x;
    const int wave = tid >> 5;

    {   // gates (PyTorch order r, z, n)
        const float ir = ws_gi[tid], iz = ws_gi[H + tid], in = ws_gi[2 * H + tid];
        const float hr = ws_gh[tid], hz = ws_gh[H + tid], hn = ws_gh[2 * H + tid];
        const float r = 1.f / (1.f + expf(-(ir + hr)));
        const float z = 1.f / (1.f + expf(-(iz + hz)));
        const float n = tanhf(in + r * hn);
        const float h = (1.f - z) * n + z * ws_hidden[tid];
        s_h[tid] = h;
        d_out[V + tid] = h;
    }
    __syncthreads();

    if (wave < 2)
        wmma_gemv_tile(w_out, s_h, b_out, s_logit, H, V, wave, false);
    __syncthreads();

    if (wave == 0) {                 // log_softmax over 29 in one wave
        float v  = (tid < V) ? s_logit[tid] : -1e30f;
        float mx = wave_max(v);
        float e  = (tid < V) ? expf(v - mx) : 0.f;
        float sm = wave_sum(e);
        if (tid < V) d_out[tid] = v - mx - logf(sm);
    }
}

// ---------------------------------------------------------------------------
extern "C" void kernel_launch(void* const* d_in, const int* in_sizes, int n_in,
                              void* d_out, int out_size, void* d_ws, size_t ws_size,
                              hipStream_t stream) {
    const int*   input_tensor = (const int*)  d_in[0];
    const float* hidden       = (const float*)d_in[1];
    const float* enc          = (const float*)d_in[2];
    const int*   cond         = (const int*)  d_in[3];
    const int*   is_head      = (const int*)  d_in[4];
    const float* emb          = (const float*)d_in[5];
    const float* w_l2d        = (const float*)d_in[6];
    const float* b_l2d        = (const float*)d_in[7];
    const float* w_attn       = (const float*)d_in[8];
    const float* b_attn       = (const float*)d_in[9];
    const float* w_comb       = (const float*)d_in[10];
    const float* b_comb       = (const float*)d_in[11];
    const float* w_ih         = (const float*)d_in[12];
    const float* w_hh         = (const float*)d_in[13];
    const float* b_ih         = (const float*)d_in[14];
    const float* b_hh         = (const float*)d_in[15];
    const float* w_out        = (const float*)d_in[16];
    const float* b_out        = (const float*)d_in[17];

    float* out = (float*)d_out;           // 29 logits | 256 h_new | 20 attn
    float* ws  = (float*)d_ws;
    float* ws_hidden = ws;                // 256
    float* ws_xcomb  = ws + 256;          // 512
    float* ws_outp   = ws + 768;          // 256
    float* ws_gh     = ws + 1024;         // 768
    float* ws_gi     = ws + 1792;         // 768  (total 2560 floats = 10 KB)

    k1_front<<<1, 256, 0, stream>>>(input_tensor, hidden, enc, cond, is_head,
                                    emb, w_l2d, b_l2d, w_attn, b_attn,
                                    ws_hidden, ws_xcomb, out + V + H);
    k2_comb_gh<<<64, 32, 0, stream>>>(w_comb, b_comb, w_hh, b_hh,
                                      ws_xcomb, ws_hidden, ws_outp, ws_gh);
    k3_gi<<<48, 32, 0, stream>>>(w_ih, b_ih, ws_outp, ws_gi);
    k4_tail<<<1, 256, 0, stream>>>(ws_gi, ws_gh, ws_hidden, w_out, b_out, out);
}